// FasterRCNN_62775241998752
// MI455X (gfx1250) — compile-verified
//
#include <hip/hip_runtime.h>

typedef unsigned int u32;
typedef unsigned long long u64;
typedef __attribute__((ext_vector_type(4))) u32 v4u32;
typedef __attribute__((ext_vector_type(4))) int  v4i32;
typedef __attribute__((ext_vector_type(8))) int  v8i32;

#define TILE 256
#define MGT  64
#define BIGF 1.0e18f
#define NEUTRALF (-100000000.0f)

__global__ __launch_bounds__(256) void rcnn_match_kernel(
    const float* __restrict__ anchors,   // (B, N, 4) xyxy
    const float* __restrict__ gt,        // (B, 64, 5) xyxy+cls
    float* __restrict__ out,             // (B, N, 5)
    int N)
{
    __shared__ float  sgt_raw[MGT * 5];   // TDM landing buffer (1280 B)
    __shared__ float4 sbox[MGT];          // preprocessed boxes
    __shared__ float  sarea[MGT];
    __shared__ float  scls[MGT];
    __shared__ float  sout[TILE * 5];     // output staging for coalescing

    const int tid = threadIdx.x;
    const int b   = blockIdx.y;
    const int a0  = blockIdx.x * TILE;

    // ---- Tensor Data Mover: async DMA of this image's GT block (320 f32,
    //      contiguous) into LDS. Issued once, by wave 0 only. ----
    if (tid < 32) {
        u64 gaddr = (u64)(const void*)(gt + (u64)b * (MGT * 5));
        u32 lds   = (u32)(u64)(&sgt_raw[0]);   // low 32 bits = LDS byte offset
        v4u32 g0;
        g0.x = 1u;                                        // count=1 (valid D#)
        g0.y = lds;                                       // lds_addr
        g0.z = (u32)(gaddr & 0xffffffffu);                // global_addr[31:0]
        g0.w = (u32)((gaddr >> 32) & 0x01ffffffu)
             | (2u << 30);                                // addr[56:32] | type=2
        v8i32 g1;
        g1[0] = (int)(2u << 16);                          // data_size=4B, mask=0
        g1[1] = (int)((u32)(MGT * 5) << 16);              // tensor_dim0 lo16
        g1[2] = (int)(1u << 16);                          // dim0 hi=0 | tensor_dim1=1
        g1[3] = (int)((u32)(MGT * 5) << 16);              // dim1 hi=0 | tile_dim0=320
        g1[4] = 1;                                        // tile_dim1=1, tile_dim2=0
        g1[5] = MGT * 5;                                  // tensor_dim0_stride lo32
        g1[6] = 0;
        g1[7] = 0;
        v4i32 gz = {0, 0, 0, 0};
#if __clang_major__ >= 23
        v8i32 gz8 = {0, 0, 0, 0, 0, 0, 0, 0};
        __builtin_amdgcn_tensor_load_to_lds(g0, g1, gz, gz, gz8, 0);
#else
        __builtin_amdgcn_tensor_load_to_lds(g0, g1, gz, gz, 0);
#endif
        __builtin_amdgcn_s_wait_tensorcnt(0);
    }
    __syncthreads();

    // ---- Prep: neutralize padded GT rows (cls==-1) with a degenerate
    //      far-away box + zero area. Removes per-iteration validity selects:
    //      such boxes always yield IoU 0, which can only win when the anchor
    //      is background (quality <= 0.4) where the index is irrelevant. ----
    if (tid < MGT) {
        float x0 = sgt_raw[tid * 5 + 0];
        float y0 = sgt_raw[tid * 5 + 1];
        float x1 = sgt_raw[tid * 5 + 2];
        float y1 = sgt_raw[tid * 5 + 3];
        float cc = sgt_raw[tid * 5 + 4];
        bool  v  = (cc != -1.0f);
        float4 bx;
        bx.x = v ? x0 : BIGF;
        bx.y = v ? y0 : BIGF;
        bx.z = v ? x1 : BIGF;
        bx.w = v ? y1 : BIGF;
        sbox[tid]  = bx;
        sarea[tid] = v ? (x1 - x0) * (y1 - y0) : 0.0f;
        scls[tid]  = cc;
    }
    __syncthreads();

    // ---- Main IoU loop: one anchor per thread. Best tracked as a fraction
    //      (bi/bu) and compared by cross-multiplication (all terms >= 0),
    //      so no division inside the loop. Strict '>' replicates
    //      argmax-first-index tie-breaking. ----
    const int a  = a0 + tid;
    const int ac = (a < N) ? a : (N - 1);
    const float4 A = ((const float4*)anchors)[(u64)b * N + ac];
    const float aa = (A.z - A.x) * (A.w - A.y);

    float bi = -1.0f;   // numerator of best IoU (init represents -1)
    float bu = 1.0f;    // denominator of best IoU
    int   bidx = 0;

#pragma unroll 8
    for (int m = 0; m < MGT; ++m) {
        float4 G = sbox[m];
        float  ga = sarea[m];
        float xl = fminf(A.z, G.z) - fmaxf(A.x, G.x);
        float yl = fminf(A.w, G.w) - fmaxf(A.y, G.y);
        float inter = ((xl > 0.0f) && (yl > 0.0f)) ? xl * yl : 0.0f;
        float uni = aa + ga - inter;
        float den = (uni == 0.0f) ? 1.0f : uni;   // reference's union==0 guard
        bool upd = (inter * bu) > (bi * den);     // iou_m > iou_best
        bi   = upd ? inter : bi;
        bu   = upd ? den   : bu;
        bidx = upd ? m     : bidx;
    }

    // ---- Thresholds (single division, identical formula to reference) ----
    float q = bi / bu;
    float4 Gb = sbox[bidx];
    float  c  = scls[bidx];
    float o0, o1, o2, o3, o4;
    if (q >= 0.6f)      { o0 = Gb.x; o1 = Gb.y; o2 = Gb.z; o3 = Gb.w; o4 = c; }
    else if (q > 0.4f)  { o0 = o1 = o2 = o3 = o4 = NEUTRALF; }
    else                { o0 = o1 = o2 = o3 = o4 = -1.0f; }

    // stride-5 write: banks (5*tid+k) mod 64 all distinct -> conflict-free
    sout[tid * 5 + 0] = o0;
    sout[tid * 5 + 1] = o1;
    sout[tid * 5 + 2] = o2;
    sout[tid * 5 + 3] = o3;
    sout[tid * 5 + 4] = o4;
    __syncthreads();

    // ---- Coalesced tile writeback: consecutive lanes -> consecutive dwords ----
    const u64 base  = ((u64)b * N + (u64)a0) * 5ull;
    const int nanch = (N - a0 < TILE) ? (N - a0) : TILE;
    const int limit = nanch * 5;
    for (int i = tid; i < limit; i += TILE)
        out[base + i] = sout[i];
}

extern "C" void kernel_launch(void* const* d_in, const int* in_sizes, int n_in,
                              void* d_out, int out_size, void* d_ws, size_t ws_size,
                              hipStream_t stream) {
    const float* anchors = (const float*)d_in[0];   // (B, N, 4) f32
    const float* gt      = (const float*)d_in[1];   // (B, 64, 5) f32
    float* out = (float*)d_out;                     // (B, N, 5) f32

    const int B = in_sizes[1] / (MGT * 5);
    const int N = in_sizes[0] / (B * 4);

    dim3 grid((N + TILE - 1) / TILE, B);
    rcnn_match_kernel<<<grid, TILE, 0, stream>>>(anchors, gt, out, N);
}